// GraphConvolution_76390288327212
// MI455X (gfx1250) — compile-verified
//
#include <hip/hip_runtime.h>

typedef __attribute__((ext_vector_type(2))) float v2f;
typedef __attribute__((ext_vector_type(8))) float v8f;

#define DIN  512
#define DOUT 256

// ---------------------------------------------------------------------------
// Kernel 1: pre_sup[M,256] = x[M,512] @ W[512,256] via V_WMMA_F32_16X16X4_F32
// Block = 128 threads (4 waves). Block tile: 32 rows x 128 cols.
// Each wave owns a 32x32 register tile = 2 M-tiles x 2 N-tiles (4 acc),
// so each A fragment feeds 2 WMMAs and each B fragment feeds 2 WMMAs:
// per K-step (K+=4): 2 ds_load_b64 + 4 global_load_b32 -> 4 WMMAs
// (1.5 memory instructions per WMMA).
// Fragment layouts per CDNA5 ISA 7.12.2:
//   A (16x4, 2 VGPRs): lanes 0-15 -> M=lane, reg0=K0,reg1=K1; lanes16-31 -> K2,K3
//   B (4x16, 2 VGPRs): lanes 0-15 -> N=lane, reg0=K0,reg1=K1; lanes16-31 -> K2,K3
//   C (16x16, 8 VGPRs): reg i -> row i (lanes 0-15) / row i+8 (lanes 16-31)
// ---------------------------------------------------------------------------
__global__ __launch_bounds__(128)
void gemm_wmma_f32(const float* __restrict__ x, const float* __restrict__ W,
                   float* __restrict__ pre, int M)
{
    __shared__ float As[32][DIN + 4];   // +4 pad: row stride 516 % 64 banks = 4

    const int m0   = blockIdx.x * 32;
    const int wave = threadIdx.x >> 5;
    const int lane = threadIdx.x & 31;
    const int half = lane >> 4;         // 0: lanes 0-15, 1: lanes 16-31
    const int lm   = lane & 15;
    const int n0   = blockIdx.y * 128 + wave * 32;
    const bool full = (m0 + 32 <= M);   // uniform: always true when 32|M

    // Cooperative, coalesced load of the 32x512 A tile (float4 granularity).
    if (full) {
        for (int i = threadIdx.x; i < 32 * (DIN / 4); i += 128) {
            const int r = i / (DIN / 4);
            const int c = (i % (DIN / 4)) * 4;
            *(float4*)&As[r][c] =
                *(const float4*)(x + (size_t)(m0 + r) * DIN + c);
        }
    } else {
        for (int i = threadIdx.x; i < 32 * (DIN / 4); i += 128) {
            const int r = i / (DIN / 4);
            const int c = (i % (DIN / 4)) * 4;
            float4 v = make_float4(0.f, 0.f, 0.f, 0.f);
            if (m0 + r < M)
                v = *(const float4*)(x + (size_t)(m0 + r) * DIN + c);
            *(float4*)&As[r][c] = v;
        }
    }
    __syncthreads();

    v8f acc00 = {};   // rows m0+ 0..15, cols n0+ 0..15
    v8f acc01 = {};   // rows m0+ 0..15, cols n0+16..31
    v8f acc10 = {};   // rows m0+16..31, cols n0+ 0..15
    v8f acc11 = {};   // rows m0+16..31, cols n0+16..31
    #pragma unroll 4
    for (int k = 0; k < DIN; k += 4) {
        const int ka = k + 2 * half;                 // this half-wave's K pair
        v2f a0 = *(const v2f*)&As[lm     ][ka];      // ds_load_b64
        v2f a1 = *(const v2f*)&As[lm + 16][ka];      // ds_load_b64
        const float* wrow0 = W + (size_t)(ka + 0) * DOUT + n0 + lm;
        const float* wrow1 = W + (size_t)(ka + 1) * DOUT + n0 + lm;
        v2f b0, b1;
        b0.x = wrow0[0];  b0.y = wrow1[0];           // cols n0+lm
        b1.x = wrow0[16]; b1.y = wrow1[16];          // cols n0+16+lm
        acc00 = __builtin_amdgcn_wmma_f32_16x16x4_f32(
                    false, a0, false, b0, (short)0, acc00, false, false);
        acc01 = __builtin_amdgcn_wmma_f32_16x16x4_f32(
                    false, a0, false, b1, (short)0, acc01, false, false);
        acc10 = __builtin_amdgcn_wmma_f32_16x16x4_f32(
                    false, a1, false, b0, (short)0, acc10, false, false);
        acc11 = __builtin_amdgcn_wmma_f32_16x16x4_f32(
                    false, a1, false, b1, (short)0, acc11, false, false);
    }

    float* dst0 = pre + (size_t)(m0 +      8 * half) * DOUT + n0 + lm;
    float* dst1 = pre + (size_t)(m0 + 16 + 8 * half) * DOUT + n0 + lm;
    if (full) {
        #pragma unroll
        for (int i = 0; i < 8; ++i) {
            dst0[(size_t)i * DOUT +  0] = acc00[i];
            dst0[(size_t)i * DOUT + 16] = acc01[i];
            dst1[(size_t)i * DOUT +  0] = acc10[i];
            dst1[(size_t)i * DOUT + 16] = acc11[i];
        }
    } else {
        #pragma unroll
        for (int i = 0; i < 8; ++i) {
            if (m0 + 8 * half + i < M) {
                dst0[(size_t)i * DOUT +  0] = acc00[i];
                dst0[(size_t)i * DOUT + 16] = acc01[i];
            }
            if (m0 + 16 + 8 * half + i < M) {
                dst1[(size_t)i * DOUT +  0] = acc10[i];
                dst1[(size_t)i * DOUT + 16] = acc11[i];
            }
        }
    }
}

// ---------------------------------------------------------------------------
// Kernel 2: zero the accumulator (d_out)
// ---------------------------------------------------------------------------
__global__ void zero_f4(float4* __restrict__ p, int n4)
{
    int i = blockIdx.x * blockDim.x + threadIdx.x;
    if (i < n4) p[i] = make_float4(0.f, 0.f, 0.f, 0.f);
}

// ---------------------------------------------------------------------------
// Kernel 3: SpMM scatter. One wave (32 lanes) per edge; each lane handles 8
// contiguous floats of the 256-wide row: 2x float4 gather from pre_sup
// (102 MB -> resident in 192 MB L2) + 8 global_atomic_add_f32 into out.
// ---------------------------------------------------------------------------
__global__ __launch_bounds__(256)
void spmm_scatter(const float* __restrict__ pre, const float* __restrict__ vals,
                  const int* __restrict__ rows, const int* __restrict__ cols,
                  float* __restrict__ out, int E)
{
    const int lane = threadIdx.x & 31;
    const int e    = blockIdx.x * (blockDim.x >> 5) + (threadIdx.x >> 5);
    if (e >= E) return;

    const int   r = rows[e];   // wave-uniform
    const int   c = cols[e];
    const float v = vals[e];

    const float4* src = (const float4*)(pre + (size_t)c * DOUT) + lane * 2;
    float*        dst = out + (size_t)r * DOUT + lane * 8;

    float4 p0 = src[0];
    float4 p1 = src[1];

    unsafeAtomicAdd(dst + 0, v * p0.x);
    unsafeAtomicAdd(dst + 1, v * p0.y);
    unsafeAtomicAdd(dst + 2, v * p0.z);
    unsafeAtomicAdd(dst + 3, v * p0.w);
    unsafeAtomicAdd(dst + 4, v * p1.x);
    unsafeAtomicAdd(dst + 5, v * p1.y);
    unsafeAtomicAdd(dst + 6, v * p1.z);
    unsafeAtomicAdd(dst + 7, v * p1.w);
}

// ---------------------------------------------------------------------------
// Kernel 4: out = relu(out + bias), vectorized float4 (DOUT=256 -> 64 float4/row)
// ---------------------------------------------------------------------------
__global__ void bias_relu(float4* __restrict__ out, const float* __restrict__ bias,
                          int n4)
{
    int i = blockIdx.x * blockDim.x + threadIdx.x;
    if (i >= n4) return;
    const int cb = (i & 63) * 4;        // column of first element
    float4 o = out[i];
    o.x = fmaxf(o.x + bias[cb + 0], 0.f);
    o.y = fmaxf(o.y + bias[cb + 1], 0.f);
    o.z = fmaxf(o.z + bias[cb + 2], 0.f);
    o.w = fmaxf(o.w + bias[cb + 3], 0.f);
    out[i] = o;
}

// ---------------------------------------------------------------------------
extern "C" void kernel_launch(void* const* d_in, const int* in_sizes, int n_in,
                              void* d_out, int out_size, void* d_ws, size_t ws_size,
                              hipStream_t stream)
{
    const float* x    = (const float*)d_in[0];
    const float* W    = (const float*)d_in[1];
    const float* bias = (const float*)d_in[2];
    const float* vals = (const float*)d_in[3];
    const int*   rows = (const int*)d_in[4];
    const int*   cols = (const int*)d_in[5];

    const int M = in_sizes[0] / DIN;     // 100000
    const int E = in_sizes[3];           // 3200000

    float* pre = (float*)d_ws;           // needs M*DOUT*4 = 102.4 MB scratch
    float* out = (float*)d_out;

    // 1) GEMM: grid 3125 x 2, 128 threads (4 waves, each a 32x32 register tile)
    dim3 gg((M + 31) / 32, DOUT / 128);
    hipLaunchKernelGGL(gemm_wmma_f32, gg, dim3(128), 0, stream, x, W, pre, M);

    // 2) zero out
    const int n4 = (M * DOUT) / 4;
    hipLaunchKernelGGL(zero_f4, dim3((n4 + 255) / 256), dim3(256), 0, stream,
                       (float4*)out, n4);

    // 3) SpMM scatter: 8 edges per 256-thread block
    const int edges_per_block = 256 / 32;
    hipLaunchKernelGGL(spmm_scatter,
                       dim3((E + edges_per_block - 1) / edges_per_block),
                       dim3(256), 0, stream, pre, vals, rows, cols, out, E);

    // 4) bias + relu
    hipLaunchKernelGGL(bias_relu, dim3((n4 + 255) / 256), dim3(256), 0, stream,
                       (float4*)out, bias, n4);
}